// ChaoticNet_5961414607652
// MI455X (gfx1250) — compile-verified
//
#include <hip/hip_runtime.h>
#include <math.h>
#include <stdint.h>

typedef float v2f __attribute__((ext_vector_type(2)));
typedef float v8f __attribute__((ext_vector_type(8)));

#define TRAJ_LEN 1000
#define TRAJ_PAD 1024

// GEMM tiling: block = 64(M) x 128(N), 8 waves, each wave 64x16. KC=16 double-buffered.
#define BM   64
#define BN   128
#define KC   16
#define APAD 20    // 80B row stride: 16B-multiple, lane banks l16*20 mod 64 all distinct
#define BPAD 136   // 544B row stride: 16B-multiple, 2*136 mod 64 = 16 -> K-halves use disjoint banks

// ---------------------------------------------------------------------------
// Phase 1: serial tent-map trajectory + exclusive prefix tables (1 thread).
// ---------------------------------------------------------------------------
__global__ void traj_kernel(const float* __restrict__ ic_p,
                            const float* __restrict__ th_p,
                            float* __restrict__ traj,
                            float* __restrict__ cgt,
                            float* __restrict__ csq,
                            float* __restrict__ cent) {
    if (threadIdx.x != 0) return;
    const float th = *th_p;
    float c = *ic_p;
    float sgt = 0.0f, ssq = 0.0f, sce = 0.0f;
    for (int i = 0; i < TRAJ_LEN; ++i) {
        traj[i] = c;
        cgt[i]  = sgt;
        csq[i]  = ssq;
        cent[i] = sce;
        sgt += (c > 0.5f) ? 1.0f : 0.0f;
        ssq += c * c;
        sce += c * log2f(c + 1e-10f);
        c = (c < th) ? (c / th) : ((1.0f - c) / (1.0f - th));
    }
}

// ---------------------------------------------------------------------------
// Phase 2: first-min argmin over the trajectory + feature gather.
// Trajectory + prefix tables staged in LDS; uniform-address ds_load broadcast.
// ---------------------------------------------------------------------------
__global__ void feat_kernel(const float* __restrict__ x,
                            const float* __restrict__ traj,
                            const float* __restrict__ cgt,
                            const float* __restrict__ csq,
                            const float* __restrict__ cent,
                            float* __restrict__ feats) {
    __shared__ float s_traj[TRAJ_PAD];
    __shared__ float s_cgt[TRAJ_PAD];
    __shared__ float s_csq[TRAJ_PAD];
    __shared__ float s_cent[TRAJ_PAD];
    for (int i = threadIdx.x; i < TRAJ_PAD; i += blockDim.x) {
        int j = (i < TRAJ_LEN) ? i : (TRAJ_LEN - 1);
        s_traj[i] = traj[j];
        s_cgt[i]  = cgt[j];
        s_csq[i]  = csq[j];
        s_cent[i] = cent[j];
    }
    __syncthreads();

    const int gid = blockIdx.x * blockDim.x + threadIdx.x;
    const float xv = x[gid];

    float best = fabsf(xv - s_traj[0]);
    int   idx  = 0;
#pragma unroll 4
    for (int i = 1; i < TRAJ_LEN; ++i) {
        float d = fabsf(xv - s_traj[i]);     // strict < keeps FIRST min index
        if (d < best) { best = d; idx = i; }
    }

    const float tt      = (float)idx;
    const float ttss    = (idx > 0) ? (s_cgt[idx] / fmaxf(tt, 1.0f)) : 0.0f;
    const float energy  = s_csq[idx];
    const float entropy = -s_cent[idx];

    float4 f4 = make_float4(ttss, energy, tt, entropy);
    *(float4*)(feats + (size_t)gid * 4) = f4;
}

// ---------------------------------------------------------------------------
// Async global -> LDS copy, 16B per lane, tracked by ASYNCcnt (gfx1250).
// VDST = VGPR holding the wave-relative LDS byte address (low 32 bits of the
// generic shared pointer), VADDR = 64-bit global address.
// ---------------------------------------------------------------------------
__device__ __forceinline__ void async_b128_to_lds(const float* g, const float* l) {
    asm volatile("global_load_async_to_lds_b128 %0, %1, off"
                 :: "v"((uint32_t)(uintptr_t)l), "v"((uint64_t)(uintptr_t)g)
                 : "memory");
}
__device__ __forceinline__ void wait_asynccnt0() {
    asm volatile("s_wait_asynccnt 0x0" ::: "memory");
}

// ---------------------------------------------------------------------------
// Phase 3/4: f32 WMMA GEMM, C = relu?(A[MxK] * B[KxN] + bias[N]).
// Block tile 64x128, double-buffered async LDS staging, V_WMMA_F32_16X16X4_F32.
//
// Per-lane WMMA layouts (ISA 7.12.2, wave32):
//   A 16x4 : lanes 0-15 row M=lane, v0/v1 = K 0/1; lanes 16-31 K 2/3.
//   B 4x16 : lanes 0-15 col N=lane, v0/v1 = K 0/1; lanes 16-31 K 2/3.
//   C 16x16: VGPR r = row (r + 8*(lane>=16)), col = lane&15.
// ---------------------------------------------------------------------------
__global__ void __launch_bounds__(256)
wmma_gemm_kernel(const float* __restrict__ A, const float* __restrict__ B,
                 const float* __restrict__ bias, float* __restrict__ C,
                 int M, int N, int K, int relu) {
    __shared__ __align__(16) float sA[2][BM][APAD];
    __shared__ __align__(16) float sB[2][KC][BPAD];

    const int t    = threadIdx.x;
    const int wave = t >> 5;
    const int lane = t & 31;
    const int half = lane >> 4;
    const int l16  = lane & 15;

    const int nblk = N / BN;
    const int m0 = (blockIdx.x / nblk) * BM;
    const int n0 = (blockIdx.x % nblk) * BN;

    // staging assignment: A slab 64x16 -> 256 threads x float4; B slab 16x128 -> 2 rounds
    const int arow = t >> 2;
    const int aseg = (t & 3) * 4;

    auto stage = [&](int buf, int kc) {
        async_b128_to_lds(A + (size_t)(m0 + arow) * K + kc + aseg, &sA[buf][arow][aseg]);
#pragma unroll
        for (int r = 0; r < 2; ++r) {
            int unit = t + r * 256;
            int row  = unit >> 5;
            int seg  = (unit & 31) * 4;
            async_b128_to_lds(B + (size_t)(kc + row) * N + n0 + seg, &sB[buf][row][seg]);
        }
    };

    const int nl = wave * 16 + l16;          // column within block tile

    v8f c0 = {}, c1 = {}, c2 = {}, c3 = {};

    stage(0, 0);
    wait_asynccnt0();
    __syncthreads();

    for (int kc = 0; kc < K; kc += KC) {
        const int  cur  = (kc >> 4) & 1;
        const bool more = (kc + KC) < K;
        if (more) stage(cur ^ 1, kc + KC);   // async DMA overlaps the WMMAs below

#pragma unroll
        for (int ks = 0; ks < KC; ks += 4) {
            const int kk = ks + 2 * half;
            v2f b;
            b.x = sB[cur][kk][nl];
            b.y = sB[cur][kk + 1][nl];
            v2f a0 = *(const v2f*)&sA[cur][l16 +  0][kk];
            v2f a1 = *(const v2f*)&sA[cur][l16 + 16][kk];
            v2f a2 = *(const v2f*)&sA[cur][l16 + 32][kk];
            v2f a3 = *(const v2f*)&sA[cur][l16 + 48][kk];
            c0 = __builtin_amdgcn_wmma_f32_16x16x4_f32(false, a0, false, b, (short)0, c0, false, false);
            c1 = __builtin_amdgcn_wmma_f32_16x16x4_f32(false, a1, false, b, (short)0, c1, false, false);
            c2 = __builtin_amdgcn_wmma_f32_16x16x4_f32(false, a2, false, b, (short)0, c2, false, false);
            c3 = __builtin_amdgcn_wmma_f32_16x16x4_f32(false, a3, false, b, (short)0, c3, false, false);
        }

        if (more) wait_asynccnt0();          // my async issues for buffer cur^1 done
        __syncthreads();                     // everyone done: next buffer ready, cur reusable
    }

    // epilogue: bias + optional ReLU, coalesced b32 stores
    const int   n  = n0 + nl;
    const float bn = bias[n];
    v8f acc[4] = {c0, c1, c2, c3};
#pragma unroll
    for (int j = 0; j < 4; ++j) {
#pragma unroll
        for (int r = 0; r < 8; ++r) {
            int m = m0 + j * 16 + r + 8 * half;
            float v = acc[j][r] + bn;
            if (relu) v = fmaxf(v, 0.0f);
            C[(size_t)m * N + n] = v;
        }
    }
}

// ---------------------------------------------------------------------------
// Launch: traj -> features -> GEMM1(+bias,ReLU) -> GEMM2(+bias)
// Inputs (setup_inputs order): x, W1, b1, W2, b2, initial_cond, threshold, eps
// ---------------------------------------------------------------------------
extern "C" void kernel_launch(void* const* d_in, const int* in_sizes, int n_in,
                              void* d_out, int out_size, void* d_ws, size_t ws_size,
                              hipStream_t stream) {
    (void)in_sizes; (void)n_in; (void)out_size; (void)ws_size;

    const float* x  = (const float*)d_in[0];   // [512,256]
    const float* W1 = (const float*)d_in[1];   // [1024,2048]
    const float* b1 = (const float*)d_in[2];   // [2048]
    const float* W2 = (const float*)d_in[3];   // [2048,512]
    const float* b2 = (const float*)d_in[4];   // [512]
    const float* ic = (const float*)d_in[5];
    const float* th = (const float*)d_in[6];
    float* out = (float*)d_out;                // [512,512]

    const int B = 512, N = 256, F = 1024, H = 2048, O = 512;

    float* traj  = (float*)d_ws;               // 1024
    float* cgt   = traj + TRAJ_PAD;            // 1024
    float* csq   = cgt  + TRAJ_PAD;            // 1024
    float* cent  = csq  + TRAJ_PAD;            // 1024
    float* feats = cent + TRAJ_PAD;            // 512*1024
    float* hbuf  = feats + (size_t)B * F;      // 512*2048

    traj_kernel<<<1, 32, 0, stream>>>(ic, th, traj, cgt, csq, cent);

    feat_kernel<<<(B * N) / 256, 256, 0, stream>>>(x, traj, cgt, csq, cent, feats);

    {   // h = relu(feats @ W1 + b1) : M=512, K=1024, N=2048
        int blocks = (B / BM) * (H / BN);      // 8*16 = 128
        wmma_gemm_kernel<<<blocks, 256, 0, stream>>>(feats, W1, b1, hbuf, B, H, F, 1);
    }
    {   // out = h @ W2 + b2 : M=512, K=2048, N=512
        int blocks = (B / BM) * (O / BN);      // 8*4 = 32
        wmma_gemm_kernel<<<blocks, 256, 0, stream>>>(hbuf, W2, b2, out, B, O, H, 0);
    }
}